// PPO_82325933129955
// MI455X (gfx1250) — compile-verified
//
#include <hip/hip_runtime.h>
#include <hip/hip_bf16.h>
#include <math.h>

typedef __attribute__((ext_vector_type(2))) float v2f;
typedef __attribute__((ext_vector_type(8))) float v8f;

#define DEVFN static __device__ __forceinline__

DEVFN v8f wmma4(v2f a, v2f b, v8f c) {
  // (neg_a, A, neg_b, B, c_mod, C, reuse_a, reuse_b)
  return __builtin_amdgcn_wmma_f32_16x16x4_f32(false, a, false, b, (short)0, c,
                                               false, false);
}

DEVFN float sigmoidf_(float v) { return 1.f / (1.f + expf(-v)); }

// ---------------------------------------------------------------------------
// Kernel 0: zero the 64-word feature-max workspace (harness poisons d_ws).
// ---------------------------------------------------------------------------
__global__ void ppo_zero_feat(unsigned* p) { p[threadIdx.x] = 0u; }

// ---------------------------------------------------------------------------
// Kernel 1: point encoder.  h3 = relu(W3 relu(W2 relu(W1 p + b1) + b2) + b3),
// feat = max over points.  WMMA f32 16x16x4, 16 points per tile per wave.
//
// Layout trick: layer-n D output (vgpr r = feature r | r+8 across lane
// halves) is fed *directly* as layer-(n+1) B fragments in register pairs
// (d[2kc], d[2kc+1]); the A fragments are built with the matching permuted
// weight columns, so no cross-lane data movement is needed between layers.
// K-slot feature map for chunk kc, vgpr j, lane-half h:
//   layer2 B from d1:  feat = 2*kc + j + 8*h            (kc 0..3)
//   layer3 B from d2:  feat = 16*(kc>>2) + 2*(kc&3) + j + 8*h   (kc 0..7)
// ---------------------------------------------------------------------------
__global__ void __launch_bounds__(256) ppo_encoder(
    const float* __restrict__ x, int N,
    const float* __restrict__ c1w, const float* __restrict__ c1b,
    const float* __restrict__ c2w, const float* __restrict__ c2b,
    const float* __restrict__ c3w, const float* __restrict__ c3b,
    unsigned* __restrict__ gfeat)
{
  __shared__ unsigned smax[64];
  const int tid = threadIdx.x;
  if (tid < 64) smax[tid] = 0u;
  __syncthreads();

  const float* __restrict__ pts0 = x + 3;
  const float* __restrict__ pts1 = x + 3 + N;

  const int lane = tid & 31;
  const int half = lane >> 4;   // 0: lanes 0-15, 1: lanes 16-31
  const int lm   = lane & 15;   // M row within a 16-row tile

  // ---- A fragments ----
  // layer 1 (real K=2, slots at half=1 are zero padding)
  v2f a1;
  {
    float w0 = c1w[lm * 2 + 0], w1 = c1w[lm * 2 + 1];
    a1[0] = half ? 0.f : w0;
    a1[1] = half ? 0.f : w1;
  }
  // layer 2: permuted columns to match d1-register B fragments
  v2f a2[2][4];
  #pragma unroll
  for (int t2 = 0; t2 < 2; ++t2)
    #pragma unroll
    for (int kc = 0; kc < 4; ++kc) {
      int row = 16 * t2 + lm;
      a2[t2][kc][0] = c2w[row * 16 + (2 * kc + 0 + 8 * half)];
      a2[t2][kc][1] = c2w[row * 16 + (2 * kc + 1 + 8 * half)];
    }
  // layer 3: permuted columns to match d2-register B fragments
  v2f a3[4][8];
  #pragma unroll
  for (int t3 = 0; t3 < 4; ++t3)
    #pragma unroll
    for (int kc = 0; kc < 8; ++kc) {
      int row = 16 * t3 + lm;
      int col = 16 * (kc >> 2) + 2 * (kc & 3) + 8 * half;
      a3[t3][kc][0] = c3w[row * 32 + col + 0];
      a3[t3][kc][1] = c3w[row * 32 + col + 1];
    }

  // ---- bias C fragments (vgpr r = M r | M r+8) ----
  float bias1[8], bias2[2][8], bias3[4][8];
  #pragma unroll
  for (int r = 0; r < 8; ++r) bias1[r] = c1b[8 * half + r];
  #pragma unroll
  for (int t2 = 0; t2 < 2; ++t2)
    #pragma unroll
    for (int r = 0; r < 8; ++r) bias2[t2][r] = c2b[16 * t2 + 8 * half + r];
  #pragma unroll
  for (int t3 = 0; t3 < 4; ++t3)
    #pragma unroll
    for (int r = 0; r < 8; ++r) bias3[t3][r] = c3b[16 * t3 + 8 * half + r];

  float vmax[4][8];
  #pragma unroll
  for (int t3 = 0; t3 < 4; ++t3)
    #pragma unroll
    for (int r = 0; r < 8; ++r) vmax[t3][r] = 0.f;   // relu output >= 0

  const int ntiles = N >> 4;
  const int wid = blockIdx.x * (blockDim.x >> 5) + (tid >> 5);
  const int nw  = gridDim.x * (blockDim.x >> 5);

  for (int tile = wid; tile < ntiles; tile += nw) {
    const int p0 = tile << 4;
    // B1: 2x16 point tile padded to 4x16 (half=1 K-slots are zero)
    float px = pts0[p0 + lm], py = pts1[p0 + lm];
    v2f b1;
    b1[0] = half ? 0.f : px;
    b1[1] = half ? 0.f : py;

    // ---- layer 1 ----
    v8f acc1;
    #pragma unroll
    for (int r = 0; r < 8; ++r) acc1[r] = bias1[r];
    acc1 = wmma4(a1, b1, acc1);
    float d1[8];
    #pragma unroll
    for (int r = 0; r < 8; ++r) d1[r] = fmaxf(acc1[r], 0.f);

    // ---- layer 2 (32x16 = two M-tiles, K=16 = 4 chunks) ----
    // B fragments are d1 register pairs (no shuffles).
    float d2[2][8];
    #pragma unroll
    for (int t2 = 0; t2 < 2; ++t2) {
      v8f acc;
      #pragma unroll
      for (int r = 0; r < 8; ++r) acc[r] = bias2[t2][r];
      #pragma unroll
      for (int kc = 0; kc < 4; ++kc) {
        v2f b;
        b[0] = d1[2 * kc + 0];
        b[1] = d1[2 * kc + 1];
        acc = wmma4(a2[t2][kc], b, acc);
      }
      #pragma unroll
      for (int r = 0; r < 8; ++r) d2[t2][r] = fmaxf(acc[r], 0.f);
    }

    // ---- layer 3 (64x16 = four M-tiles, K=32 = 8 chunks) ----
    // B fragments are d2 register pairs (no shuffles).
    #pragma unroll
    for (int t3 = 0; t3 < 4; ++t3) {
      v8f acc;
      #pragma unroll
      for (int r = 0; r < 8; ++r) acc[r] = bias3[t3][r];
      #pragma unroll
      for (int kc = 0; kc < 8; ++kc) {
        v2f b;
        b[0] = d2[kc >> 2][2 * (kc & 3) + 0];
        b[1] = d2[kc >> 2][2 * (kc & 3) + 1];
        acc = wmma4(a3[t3][kc], b, acc);
      }
      #pragma unroll
      for (int r = 0; r < 8; ++r)
        vmax[t3][r] = fmaxf(vmax[t3][r], fmaxf(acc[r], 0.f));
    }
  }

  // ---- max across the 16 point-columns within each lane half ----
  #pragma unroll
  for (int m = 1; m <= 8; m <<= 1)
    #pragma unroll
    for (int t3 = 0; t3 < 4; ++t3)
      #pragma unroll
      for (int r = 0; r < 8; ++r)
        vmax[t3][r] = fmaxf(vmax[t3][r], __shfl_xor(vmax[t3][r], m, 32));

  if (lm == 0) {   // lanes 0 and 16 hold the reduced values
    #pragma unroll
    for (int t3 = 0; t3 < 4; ++t3)
      #pragma unroll
      for (int r = 0; r < 8; ++r) {
        int f = 16 * t3 + 8 * half + r;
        atomicMax(&smax[f], __float_as_uint(vmax[t3][r]));  // vals >= 0
      }
  }
  __syncthreads();
  if (tid < 64) atomicMax(&gfeat[tid], smax[tid]);

  // ---- scalar remainder (N % 16 points); N=500000 -> never taken ----
  if (blockIdx.x == 0) {
    int rem = N & 15;
    if (tid < rem) {
      int p = N - rem + tid;
      float x0 = pts0[p], x1 = pts1[p];
      float h1[16];
      #pragma unroll
      for (int i = 0; i < 16; ++i)
        h1[i] = fmaxf(c1w[2 * i] * x0 + c1w[2 * i + 1] * x1 + c1b[i], 0.f);
      float h2[32];
      #pragma unroll
      for (int i = 0; i < 32; ++i) {
        float s = c2b[i];
        #pragma unroll
        for (int k = 0; k < 16; ++k) s += c2w[i * 16 + k] * h1[k];
        h2[i] = fmaxf(s, 0.f);
      }
      for (int i = 0; i < 64; ++i) {
        float s = c3b[i];
        #pragma unroll
        for (int k = 0; k < 32; ++k) s += c3w[i * 32 + k] * h2[k];
        atomicMax(&gfeat[i], __float_as_uint(fmaxf(s, 0.f)));
      }
    }
  }
}

// ---------------------------------------------------------------------------
// Kernel 2: everything after the max-pool (single block, 128 threads).
// ---------------------------------------------------------------------------
__global__ void __launch_bounds__(128) ppo_head(
    const float* __restrict__ x, const float* __restrict__ hx,
    const float* __restrict__ cx,
    const float* __restrict__ fcw, const float* __restrict__ fcb,
    const float* __restrict__ o1w, const float* __restrict__ o1b,
    const float* __restrict__ o2w, const float* __restrict__ o2b,
    const float* __restrict__ pw,  const float* __restrict__ pb,
    const float* __restrict__ wih, const float* __restrict__ whh,
    const float* __restrict__ bih, const float* __restrict__ bhh,
    const float* __restrict__ q1w, const float* __restrict__ q1b,
    const float* __restrict__ q2w, const float* __restrict__ q2b,
    const float* __restrict__ aw1, const float* __restrict__ ab1,
    const float* __restrict__ aw2, const float* __restrict__ ab2,
    const float* __restrict__ ciw1, const float* __restrict__ cib1,
    const float* __restrict__ ciw2, const float* __restrict__ cib2,
    const float* __restrict__ cew1, const float* __restrict__ ceb1,
    const float* __restrict__ cew2, const float* __restrict__ ceb2,
    const unsigned* __restrict__ gfeat, float* __restrict__ out)
{
  __shared__ float feat[64], obs[64], sgl[8], pose[8], z[128], gates[512],
                   hnew[128], zp1[64], zp[64], ah[16], cih[16], ceh[16],
                   logits[5];
  const int t = threadIdx.x;

  if (t < 64) feat[t] = __uint_as_float(gfeat[t]);
  __syncthreads();

  // obs = fcw @ feat + fcb ; sigmoid(o1w @ odom + o1b)
  if (t < 64) {
    float s = fcb[t];
    for (int k = 0; k < 64; ++k) s += fcw[t * 64 + k] * feat[k];
    obs[t] = s;
  } else if (t < 72) {
    int j = t - 64;
    float s = o1b[j];
    for (int k = 0; k < 3; ++k) s += o1w[j * 3 + k] * x[k];
    sgl[j] = sigmoidf_(s);
  }
  __syncthreads();

  if (t < 8) {
    float s = o2b[t];
    for (int k = 0; k < 8; ++k) s += o2w[t * 8 + k] * sgl[k];
    pose[t] = fmaxf(s, 0.f);
    out[263 + t] = pose[t];
  }
  __syncthreads();

  // z = relu(pw @ [obs; pose] + pb)
  {
    float s = pb[t];
    for (int k = 0; k < 64; ++k) s += pw[t * 72 + k] * obs[k];
    for (int k = 0; k < 8; ++k)  s += pw[t * 72 + 64 + k] * pose[k];
    z[t] = fmaxf(s, 0.f);
  }
  __syncthreads();

  // LSTM gates (512 rows, 4 per thread)
  for (int q = 0; q < 4; ++q) {
    int g = 4 * t + q;
    float s = bih[g] + bhh[g];
    const float* wi = wih + g * 128;
    const float* wh = whh + g * 128;
    for (int k = 0; k < 128; ++k) s += wi[k] * z[k] + wh[k] * hx[k];
    gates[g] = s;
  }
  __syncthreads();

  {
    float ig = sigmoidf_(gates[t]);
    float fg = sigmoidf_(gates[128 + t]);
    float gg = tanhf(gates[256 + t]);
    float og = sigmoidf_(gates[384 + t]);
    float c  = fg * cx[t] + ig * gg;
    float h  = og * tanhf(c);
    hnew[t] = h;
    out[7 + t]   = h;   // hx2
    out[135 + t] = c;   // cx2
  }
  __syncthreads();

  if (t < 64) {
    float s = q1b[t];
    for (int k = 0; k < 128; ++k) s += q1w[t * 128 + k] * hnew[k];
    zp1[t] = fmaxf(s, 0.f);
  }
  __syncthreads();
  if (t < 64) {
    float s = q2b[t];
    for (int k = 0; k < 64; ++k) s += q2w[t * 64 + k] * zp1[k];
    zp[t] = fmaxf(s, 0.f);
  }
  __syncthreads();

  if (t < 16) {
    float s = ab1[t];
    for (int k = 0; k < 64; ++k) s += aw1[t * 64 + k] * zp[k];
    ah[t] = fmaxf(s, 0.f);
  } else if (t < 32) {
    int j = t - 16;
    float s = cib1[j];
    for (int k = 0; k < 64; ++k) s += ciw1[j * 64 + k] * zp[k];
    cih[j] = fmaxf(s, 0.f);
  } else if (t < 48) {
    int j = t - 32;
    float s = ceb1[j];
    for (int k = 0; k < 64; ++k) s += cew1[j * 64 + k] * zp[k];
    ceh[j] = fmaxf(s, 0.f);
  }
  __syncthreads();

  if (t < 5) {
    float s = ab2[t];
    for (int k = 0; k < 16; ++k) s += aw2[t * 16 + k] * ah[k];
    logits[t] = s;
  } else if (t == 5) {
    float s = cib2[0];
    for (int k = 0; k < 16; ++k) s += ciw2[k] * cih[k];
    out[5] = s;
  } else if (t == 6) {
    float s = ceb2[0];
    for (int k = 0; k < 16; ++k) s += cew2[k] * ceh[k];
    out[6] = s;
  }
  __syncthreads();

  if (t == 0) {  // stable softmax over 5 logits
    float mx = logits[0];
    for (int k = 1; k < 5; ++k) mx = fmaxf(mx, logits[k]);
    float e[5], sm = 0.f;
    for (int k = 0; k < 5; ++k) { e[k] = expf(logits[k] - mx); sm += e[k]; }
    for (int k = 0; k < 5; ++k) out[k] = e[k] / sm;
  }
}

// ---------------------------------------------------------------------------
extern "C" void kernel_launch(void* const* d_in, const int* in_sizes, int n_in,
                              void* d_out, int out_size, void* d_ws,
                              size_t ws_size, hipStream_t stream) {
  const float* x    = (const float*)d_in[0];
  const float* hx   = (const float*)d_in[1];
  const float* cx   = (const float*)d_in[2];
  const float* c1w  = (const float*)d_in[3];
  const float* c1b  = (const float*)d_in[4];
  const float* c2w  = (const float*)d_in[5];
  const float* c2b  = (const float*)d_in[6];
  const float* c3w  = (const float*)d_in[7];
  const float* c3b  = (const float*)d_in[8];
  const float* fcw  = (const float*)d_in[9];
  const float* fcb  = (const float*)d_in[10];
  const float* o1w  = (const float*)d_in[11];
  const float* o1b  = (const float*)d_in[12];
  const float* o2w  = (const float*)d_in[13];
  const float* o2b  = (const float*)d_in[14];
  const float* pw   = (const float*)d_in[15];
  const float* pb   = (const float*)d_in[16];
  const float* wih  = (const float*)d_in[17];
  const float* whh  = (const float*)d_in[18];
  const float* bih  = (const float*)d_in[19];
  const float* bhh  = (const float*)d_in[20];
  const float* q1w  = (const float*)d_in[21];
  const float* q1b  = (const float*)d_in[22];
  const float* q2w  = (const float*)d_in[23];
  const float* q2b  = (const float*)d_in[24];
  const float* aw1  = (const float*)d_in[25];
  const float* ab1  = (const float*)d_in[26];
  const float* aw2  = (const float*)d_in[27];
  const float* ab2  = (const float*)d_in[28];
  const float* ciw1 = (const float*)d_in[29];
  const float* cib1 = (const float*)d_in[30];
  const float* ciw2 = (const float*)d_in[31];
  const float* cib2 = (const float*)d_in[32];
  const float* cew1 = (const float*)d_in[33];
  const float* ceb1 = (const float*)d_in[34];
  const float* cew2 = (const float*)d_in[35];
  const float* ceb2 = (const float*)d_in[36];

  const int N = (in_sizes[0] - 3) / 2;
  unsigned* gfeat = (unsigned*)d_ws;
  float* out = (float*)d_out;

  ppo_zero_feat<<<1, 64, 0, stream>>>(gfeat);
  ppo_encoder<<<256, 256, 0, stream>>>(x, N, c1w, c1b, c2w, c2b, c3w, c3b,
                                       gfeat);
  ppo_head<<<1, 128, 0, stream>>>(x, hx, cx, fcw, fcb, o1w, o1b, o2w, o2b, pw,
                                  pb, wih, whh, bih, bhh, q1w, q1b, q2w, q2b,
                                  aw1, ab1, aw2, ab2, ciw1, cib1, ciw2, cib2,
                                  cew1, ceb1, cew2, ceb2, gfeat, out);
}